// Retina_75428215652993
// MI455X (gfx1250) — compile-verified
//
#include <hip/hip_runtime.h>

// Retina glimpse extract + write-back for MI455X (gfx1250).
//
// Reference semantics (B=16384, C=3, H=W=32, G=16):
//   l clamped to [-1,1]; coords = trunc(0.5*((l+1)*32)); start = coords - 8
//   patch[b,c,i,j] = (0<=start_r+i<32 && 0<=start_c+j<32) ? x[b,c,start_r+i,start_c+j] : 0
//   full [b,c,y,x] = x[b,c,y,x] iff (y,x) lies inside the valid window, else 0
// d_out = [ full (B*C*32*32) | patch (B*C*16*16) ] in float32.
//
// Memory-bound (≈288 MiB of traffic, 0 FLOPs) -> no WMMA; optimize pure data
// movement: LDS-staged single read of the window, b128 non-temporal stores.

typedef float f4 __attribute__((ext_vector_type(4)));

__global__ __launch_bounds__(256) void retina_glimpse_kernel(
    const float* __restrict__ x,      // [B,3,32,32]
    const float* __restrict__ l,      // [B,2]
    float* __restrict__ out_full,     // [B,3,32,32]
    float* __restrict__ out_patch)    // [B,3,16,16]
{
    const int b   = blockIdx.x;
    const int tid = threadIdx.x;

    __shared__ float sm[3 * 16 * 16];   // masked window, 3 KB

    // ---- per-sample (block-uniform) window origin ----
    float lr = l[2 * b + 0];
    float lc = l[2 * b + 1];
    lr = fminf(fmaxf(lr, -1.0f), 1.0f);
    lc = fminf(fmaxf(lc, -1.0f), 1.0f);
    // same eval order as reference: 0.5 * ((l + 1) * 32), truncating cast
    const int r0 = (int)(0.5f * ((lr + 1.0f) * 32.0f)) - 8;
    const int c0 = (int)(0.5f * ((lc + 1.0f) * 32.0f)) - 8;

    const float* __restrict__ xb = x + (size_t)b * 3072;   // 3*32*32

    // ---- stage masked 3x16x16 window into LDS (one HBM read per element) ----
    #pragma unroll
    for (int t = tid; t < 768; t += 256) {                 // 3 iterations
        const int ch  = t >> 8;
        const int rem = t & 255;
        const int i   = rem >> 4;
        const int j   = rem & 15;
        const int r   = r0 + i;
        const int c   = c0 + j;
        float v = 0.0f;
        if ((unsigned)r < 32u && (unsigned)c < 32u)
            v = xb[ch * 1024 + r * 32 + c];                // 16-lane contiguous rows
        sm[t] = v;
    }
    __syncthreads();

    // ---- patch output: straight LDS -> global, 192 x b128 NT stores ----
    if (tid < 192) {
        const f4 v = ((const f4*)sm)[tid];
        __builtin_nontemporal_store(v, (f4*)(out_patch + (size_t)b * 768) + tid);
    }

    // ---- full output: 768 x b128 NT stores (zeros outside window) ----
    f4* __restrict__ fullb = (f4*)(out_full + (size_t)b * 3072);
    #pragma unroll
    for (int q = tid; q < 768; q += 256) {                 // 3 iterations
        const int ch  = q >> 8;
        const int rem = q & 255;
        const int y   = rem >> 3;       // image row 0..31
        const int xq  = rem & 7;        // float4 column chunk 0..7
        f4 v = (f4)0.0f;
        const int i = y - r0;           // window row
        if ((unsigned)i < 16u) {
            const int jb = xq * 4 - c0; // window col of first component
            const float* __restrict__ row = sm + ch * 256 + i * 16;
            #pragma unroll
            for (int k = 0; k < 4; ++k) {
                const int j = jb + k;
                if ((unsigned)j < 16u) v[k] = row[j];
            }
        }
        __builtin_nontemporal_store(v, fullb + q);
    }
}

extern "C" void kernel_launch(void* const* d_in, const int* in_sizes, int n_in,
                              void* d_out, int out_size, void* d_ws, size_t ws_size,
                              hipStream_t stream) {
    const float* x = (const float*)d_in[0];   // [B,3,32,32]
    const float* l = (const float*)d_in[1];   // [B,2]
    const int B = in_sizes[1] / 2;            // 16384
    const size_t full_elems = (size_t)in_sizes[0];  // B*3*32*32

    float* out_full  = (float*)d_out;
    float* out_patch = (float*)d_out + full_elems;

    retina_glimpse_kernel<<<B, 256, 0, stream>>>(x, l, out_full, out_patch);
}